// SparseMoE_14456859918346
// MI455X (gfx1250) — compile-verified
//
#include <hip/hip_runtime.h>
#include <math.h>

#define T_TOKENS 8192
#define DM 1024
#define HID 4096
#define NEXP 8
#define TOPK 2
#define MTILE 32
#define HCHUNK 64
#define XS_STRIDE 1028   // 1028 % 64 == 4 -> conflict-free column reads; 1028*4 % 16 == 0
#define HS_STRIDE 68     // 68 % 64 == 4

typedef float v2f __attribute__((ext_vector_type(2)));
typedef float v8f __attribute__((ext_vector_type(8)));
typedef int   v4i __attribute__((ext_vector_type(4)));

__device__ __forceinline__ void wait_async0() {
#if __has_builtin(__builtin_amdgcn_s_wait_asynccnt)
    __builtin_amdgcn_s_wait_asynccnt(0);
#else
    asm volatile("s_wait_asynccnt 0" ::: "memory");
#endif
}

struct MoeWS {
    int   topk_idx[T_TOKENS * TOPK];
    float topk_w[T_TOKENS * TOPK];
    int   counts[NEXP];
    float prob_sum[NEXP];
    float frac_cnt[NEXP];
    int   list_tok[NEXP * T_TOKENS];
    float list_w[NEXP * T_TOKENS];
};

__global__ void moe_zero_kernel(float* __restrict__ out, MoeWS* __restrict__ ws) {
    size_t i = (size_t)blockIdx.x * blockDim.x + threadIdx.x;
    size_t n = (size_t)T_TOKENS * DM;
    if (i < n) out[i] = 0.0f;
    if (i < NEXP) {
        ws->counts[i] = 0;
        ws->prob_sum[i] = 0.0f;
        ws->frac_cnt[i] = 0.0f;
    }
}

// One wave32 per token: logits, softmax, top-2, aux-loss stats.
__global__ __launch_bounds__(256) void moe_gate_kernel(const float* __restrict__ x,
                                                       const float* __restrict__ gw,
                                                       MoeWS* __restrict__ ws) {
    int wave = (int)((blockIdx.x * blockDim.x + threadIdx.x) >> 5);
    int lane = threadIdx.x & 31;
    if (wave >= T_TOKENS) return;
    const float* xr = x + (size_t)wave * DM;

    float acc[NEXP];
#pragma unroll
    for (int e = 0; e < NEXP; e++) acc[e] = 0.0f;
    for (int d = lane; d < DM; d += 32) {
        float xv = xr[d];
        const float* g = gw + d * NEXP;
#pragma unroll
        for (int e = 0; e < NEXP; e++) acc[e] += xv * g[e];
    }
#pragma unroll
    for (int e = 0; e < NEXP; e++)
        for (int off = 16; off > 0; off >>= 1)
            acc[e] += __shfl_xor(acc[e], off, 32);

    if (lane == 0) {
        float mx = acc[0];
#pragma unroll
        for (int e = 1; e < NEXP; e++) mx = fmaxf(mx, acc[e]);
        float p[NEXP], s = 0.0f;
#pragma unroll
        for (int e = 0; e < NEXP; e++) { p[e] = __expf(acc[e] - mx); s += p[e]; }
        float inv_s = 1.0f / s;
#pragma unroll
        for (int e = 0; e < NEXP; e++) p[e] *= inv_s;

        int i0 = 0; float m0 = p[0];
#pragma unroll
        for (int e = 1; e < NEXP; e++) if (p[e] > m0) { m0 = p[e]; i0 = e; }
        int i1 = -1; float m1 = -1.0f;
#pragma unroll
        for (int e = 0; e < NEXP; e++) if (e != i0 && p[e] > m1) { m1 = p[e]; i1 = e; }

        float inv = 1.0f / (m0 + m1);
        ws->topk_idx[wave * 2 + 0] = i0;
        ws->topk_idx[wave * 2 + 1] = i1;
        ws->topk_w[wave * 2 + 0] = m0 * inv;
        ws->topk_w[wave * 2 + 1] = m1 * inv;
        atomicAdd(&ws->frac_cnt[i0], 1.0f);
#pragma unroll
        for (int e = 0; e < NEXP; e++) atomicAdd(&ws->prob_sum[e], p[e]);
    }
}

__global__ void moe_assign_kernel(MoeWS* __restrict__ ws) {
    int t = blockIdx.x * blockDim.x + threadIdx.x;
    if (t >= T_TOKENS) return;
#pragma unroll
    for (int k = 0; k < TOPK; k++) {
        int e = ws->topk_idx[t * 2 + k];
        float w = ws->topk_w[t * 2 + k];
        int slot = atomicAdd(&ws->counts[e], 1);
        ws->list_tok[e * T_TOKENS + slot] = t;
        ws->list_w[e * T_TOKENS + slot] = w;
    }
}

__global__ void moe_aux_kernel(const MoeWS* __restrict__ ws, float* __restrict__ out_aux) {
    if (threadIdx.x == 0 && blockIdx.x == 0) {
        float s = 0.0f;
        for (int e = 0; e < NEXP; e++)
            s += (ws->frac_cnt[e] * (1.0f / T_TOKENS)) * (ws->prob_sum[e] * (1.0f / T_TOKENS));
        *out_aux = (float)NEXP * s;
    }
}

// Fused expert FFN: y = gelu(x@W1+b1)@W2+b2, scatter w*y into out.
// Block = 256 threads = 8 waves, M=32 tokens, hidden chunked by 64.
__global__ __launch_bounds__(256) void moe_ffn_kernel(const float* __restrict__ x,
                                                      const float* __restrict__ w1,
                                                      const float* __restrict__ b1,
                                                      const float* __restrict__ w2,
                                                      const float* __restrict__ b2,
                                                      float* __restrict__ out,
                                                      const MoeWS* __restrict__ ws) {
    __shared__ __align__(16) float xs[MTILE * XS_STRIDE];   // ~128.5 KB
    __shared__ __align__(16) float hs[MTILE * HS_STRIDE];   // ~8.7 KB
    __shared__ int   stoks[MTILE];
    __shared__ float stw[MTILE];

    const int e = blockIdx.y;
    const int ne = ws->counts[e];
    const int base = blockIdx.x * MTILE;
    if (base >= ne) return;

    const int tid  = threadIdx.x;
    const int lane = tid & 31;
    const int wv   = tid >> 5;      // 0..7
    const int mi   = wv & 1;        // M-tile (0..1)
    const int jw   = wv >> 1;       // 0..3
    const int nl   = lane & 15;
    const int koff = (lane >= 16) ? 2 : 0;
    const int mhi  = (lane >= 16) ? 8 : 0;

    if (tid < MTILE) {
        int r = base + tid;
        stoks[tid] = (r < ne) ? ws->list_tok[e * T_TOKENS + r] : 0;
        stw[tid]   = (r < ne) ? ws->list_w[e * T_TOKENS + r] : 0.0f;
    }
    __syncthreads();

    // Gather x rows for this tile into LDS (async memory->LDS path when available).
#if __has_builtin(__builtin_amdgcn_global_load_async_to_lds_b128)
    for (int i = tid; i < (MTILE * DM) / 4; i += 256) {
        int row = (i * 4) >> 10;
        int col = (i * 4) & (DM - 1);
        const float* gsrc = x + (size_t)stoks[row] * DM + col;
        float* ldst = &xs[row * XS_STRIDE + col];
        __builtin_amdgcn_global_load_async_to_lds_b128((v4i*)gsrc, (v4i*)ldst, 0, 0);
    }
    wait_async0();
#else
    for (int i = tid; i < MTILE * DM; i += 256) {
        int row = i >> 10;
        int col = i & (DM - 1);
        xs[row * XS_STRIDE + col] = x[(size_t)stoks[row] * DM + col];
    }
#endif
    __syncthreads();   // xs tile fully resident before any wave reads it

    const float* W1e = w1 + (size_t)e * DM * HID;
    const float* W2e = w2 + (size_t)e * HID * DM;

    // Persistent D-dim accumulators: wave covers N-tiles {jw + 4t}, t=0..15.
    v8f yacc[16];
#pragma unroll
    for (int t = 0; t < 16; t++) yacc[t] = (v8f){};

    const float* xrow = xs + (mi * 16 + nl) * XS_STRIDE;
    const float* hrow = hs + (mi * 16 + nl) * HS_STRIDE;

    for (int ch = 0; ch < HID; ch += HCHUNK) {
        // ---- phase 1: h[32,64] = gelu(x[32,1024] @ W1[:,ch:ch+64] + b1) ----
        v8f c = (v8f){};
        const int hcol = ch + jw * 16 + nl;
        const float* wp = W1e + hcol;
        for (int k = 0; k < DM; k += 4) {
            v2f a, b;
            a.x = xrow[k + koff];
            a.y = xrow[k + koff + 1];
            b.x = wp[(size_t)(k + koff) * HID];
            b.y = wp[(size_t)(k + koff + 1) * HID];
            c = __builtin_amdgcn_wmma_f32_16x16x4_f32(false, a, false, b,
                                                      (short)0, c, false, false);
        }
        float b1v = b1[e * HID + hcol];
        __syncthreads();   // previous chunk's phase-2 reads of hs are complete
#pragma unroll
        for (int r = 0; r < 8; r++) {
            float v = c[r] + b1v;
            float g = 0.5f * v * (1.0f + erff(v * 0.70710678118654752f));
            hs[(mi * 16 + mhi + r) * HS_STRIDE + jw * 16 + nl] = g;
        }
        __syncthreads();

        // ---- phase 2: y[32,1024] += h[32,64] @ W2[ch:ch+64, :] ----
        for (int kk = 0; kk < HCHUNK; kk += 4) {
            v2f a;
            a.x = hrow[kk + koff];
            a.y = hrow[kk + koff + 1];
            const float* w2r0 = W2e + (size_t)(ch + kk + koff) * DM;
            const float* w2r1 = w2r0 + DM;
#pragma unroll
            for (int t = 0; t < 16; t++) {
                int ncol = (jw + 4 * t) * 16 + nl;
                v2f b;
                b.x = w2r0[ncol];
                b.y = w2r1[ncol];
                yacc[t] = __builtin_amdgcn_wmma_f32_16x16x4_f32(false, a, false, b,
                                                                (short)0, yacc[t], false, false);
            }
        }
    }

    // ---- epilogue: +b2, scale by routing weight, scatter-add ----
#pragma unroll
    for (int t = 0; t < 16; t++) {
        int ncol = (jw + 4 * t) * 16 + nl;
        float b2v = b2[e * DM + ncol];
#pragma unroll
        for (int r = 0; r < 8; r++) {
            int m = mi * 16 + mhi + r;
            float val = stw[m] * (yacc[t][r] + b2v);
            atomicAdd(&out[(size_t)stoks[m] * DM + ncol], val);
        }
    }
}

extern "C" void kernel_launch(void* const* d_in, const int* in_sizes, int n_in,
                              void* d_out, int out_size, void* d_ws, size_t ws_size,
                              hipStream_t stream) {
    const float* x      = (const float*)d_in[0];
    const float* gate_w = (const float*)d_in[1];
    const float* w1     = (const float*)d_in[2];
    const float* b1     = (const float*)d_in[3];
    const float* w2     = (const float*)d_in[4];
    const float* b2     = (const float*)d_in[5];
    float* out = (float*)d_out;
    MoeWS* ws = (MoeWS*)d_ws;

    int nzero = (T_TOKENS * DM + 255) / 256;
    moe_zero_kernel<<<nzero, 256, 0, stream>>>(out, ws);
    moe_gate_kernel<<<(T_TOKENS * 32 + 255) / 256, 256, 0, stream>>>(x, gate_w, ws);
    moe_assign_kernel<<<(T_TOKENS + 255) / 256, 256, 0, stream>>>(ws);
    moe_aux_kernel<<<1, 64, 0, stream>>>(ws, out + (size_t)T_TOKENS * DM);

    dim3 grid((T_TOKENS + MTILE - 1) / MTILE, NEXP);
    moe_ffn_kernel<<<grid, 256, 0, stream>>>(x, w1, b1, w2, b2, out, ws);
}